// SetNorm_80152679678427
// MI455X (gfx1250) — compile-verified
//
#include <hip/hip_runtime.h>
#include <hip/hip_bf16.h>

typedef __attribute__((ext_vector_type(2))) float v2f;
typedef __attribute__((ext_vector_type(8))) float v8f;

#define HAS_WMMA_F32_16X16X4 __has_builtin(__builtin_amdgcn_wmma_f32_16x16x4_f32)

constexpr int   BATCH        = 64;
constexpr int   NROW         = 2048;
constexpr int   DDIM         = 256;
constexpr float EPS          = 1e-5f;
constexpr int   CHUNKS       = 16;                       // N-chunks per batch
constexpr int   ROWS_PER_CHK = NROW / CHUNKS;            // 128
constexpr int   THREADS      = 256;                      // 8 wave32
constexpr int   F4_PER_ROW   = DDIM / 4;                 // 64
constexpr int   F4_PER_CHK   = ROWS_PER_CHK * F4_PER_ROW;// 8192
constexpr int   ITERS        = F4_PER_CHK / THREADS;     // 32

// ---------------------------------------------------------------- zero ws
__global__ void setnorm_zero_ws(float* __restrict__ ws) {
    ws[threadIdx.x] = 0.0f;   // launched with exactly 2*BATCH threads
}

// ------------------------------------------------------------- stats pass
__global__ __launch_bounds__(THREADS) void setnorm_reduce(
    const float* __restrict__ x, const int* __restrict__ lengths,
    float* __restrict__ wsum, float* __restrict__ wssq)
{
    const int b       = blockIdx.y;
    const int chunk   = blockIdx.x;
    const int len     = lengths[b];
    const int rowBase = chunk * ROWS_PER_CHK;
    if (rowBase >= len) return;                       // block-uniform skip

    const float4* __restrict__ x4 =
        reinterpret_cast<const float4*>(x) + (size_t)(b * NROW + rowBase) * F4_PER_ROW;
    const int t        = threadIdx.x;
    const int validF4  = min(len - rowBase, ROWS_PER_CHK) * F4_PER_ROW;

    float sum = 0.0f, ssq = 0.0f;
#pragma unroll
    for (int i = 0; i < ITERS; ++i) {
        const int idx = t + i * THREADS;
        if (idx < validF4) {                          // wave-uniform (row granularity)
            if (i + 4 < ITERS)
                __builtin_prefetch(x4 + idx + 4 * THREADS, 0, 0); // global_prefetch_b8
            const float4 v = x4[idx];
            sum += (v.x + v.y) + (v.z + v.w);
            ssq = fmaf(v.x, v.x, ssq);  ssq = fmaf(v.y, v.y, ssq);
            ssq = fmaf(v.z, v.z, ssq);  ssq = fmaf(v.w, v.w, ssq);
        }
    }

    const unsigned lane = threadIdx.x & 31u;
#if HAS_WMMA_F32_16X16X4
    // Cross-lane reduce of (sum, ssq) with one fp32 WMMA.
    // A (16x4): vgpr0 = sum  -> A[m][0]=sum[m],  A[m][2]=sum[m+16]
    //           vgpr1 = ssq  -> A[m][1]=ssq[m],  A[m][3]=ssq[m+16]
    // B (4x16) selector: B[k][0]=1 for even k, B[k][1]=1 for odd k
    //   -> D[m][0] = sum[m]+sum[m+16], D[m][1] = ssq[m]+ssq[m+16]
    v2f A;  A[0] = sum;  A[1] = ssq;
    v2f Bm; Bm[0] = (lane == 0u || lane == 16u) ? 1.0f : 0.0f;   // rows K=0,K=2
            Bm[1] = (lane == 1u || lane == 17u) ? 1.0f : 0.0f;   // rows K=1,K=3
    v8f C = {0.f, 0.f, 0.f, 0.f, 0.f, 0.f, 0.f, 0.f};
    v8f Dm = __builtin_amdgcn_wmma_f32_16x16x4_f32(
        /*neg_a=*/false, A, /*neg_b=*/false, Bm,
        /*c_mod=*/(short)0, C, /*reuse_a=*/false, /*reuse_b=*/false);
    // lane n keeps column n; fold the 8 row-registers, then fold lane<->lane+16
    float tot = ((Dm[0] + Dm[1]) + (Dm[2] + Dm[3])) + ((Dm[4] + Dm[5]) + (Dm[6] + Dm[7]));
    tot += __shfl_xor(tot, 16, 32);
    if (lane == 0u)      atomicAdd(&wsum[b], tot);   // column 0 == total sum
    else if (lane == 1u) atomicAdd(&wssq[b], tot);   // column 1 == total sumsq
#else
    for (int off = 16; off > 0; off >>= 1) {
        sum += __shfl_xor(sum, off, 32);
        ssq += __shfl_xor(ssq, off, 32);
    }
    if (lane == 0u) { atomicAdd(&wsum[b], sum); atomicAdd(&wssq[b], ssq); }
#endif
}

// ------------------------------------------------------------- apply pass
__global__ __launch_bounds__(THREADS) void setnorm_apply(
    const float* __restrict__ x, const int* __restrict__ lengths,
    const float* __restrict__ weights, const float* __restrict__ biases,
    const float* __restrict__ wsum, const float* __restrict__ wssq,
    float* __restrict__ out)
{
    __shared__ float s_wb[2 * DDIM];       // [0,256): weights  [256,512): biases
    const int t     = threadIdx.x;
    const int b     = blockIdx.y;
    const int chunk = blockIdx.x;

    // Stage per-feature affine params into LDS with CDNA5 async-to-LDS DMA.
    {
        const unsigned ldsW = (unsigned)(t * 4);             // byte offset in LDS
        const unsigned ldsB = (unsigned)((DDIM + t) * 4);
        const float* gw = weights + t;
        const float* gb = biases  + t;
        asm volatile("global_load_async_to_lds_b32 %0, %1, off"
                     :: "v"(ldsW), "v"(gw) : "memory");
        asm volatile("global_load_async_to_lds_b32 %0, %1, off"
                     :: "v"(ldsB), "v"(gb) : "memory");
        asm volatile("s_wait_asynccnt 0" ::: "memory");
    }
    __syncthreads();

    const int   len   = lengths[b];
    const float denom = (float)len * (float)DDIM;
    const float mean  = wsum[b] / denom;
    float var = wssq[b] / denom - mean * mean;     // exact identity w/ masked sums
    var = fmaxf(var, 0.0f);
    const float inv = 1.0f / (sqrtf(var) + EPS);

    const int c4 = t & (F4_PER_ROW - 1);           // float4 column
    const int r0 = t >> 6;                         // row phase 0..3
    const float4 w4 = reinterpret_cast<const float4*>(s_wb)[c4];
    const float4 a4 = reinterpret_cast<const float4*>(s_wb + DDIM)[c4];

    const float4* __restrict__ x4 = reinterpret_cast<const float4*>(x);
    float4*       __restrict__ o4 = reinterpret_cast<float4*>(out);
    const size_t base = (size_t)(b * NROW) * F4_PER_ROW;

#pragma unroll 4
    for (int i = 0; i < ROWS_PER_CHK / 4; ++i) {
        const int    r   = chunk * ROWS_PER_CHK + i * 4 + r0;
        const size_t idx = base + (size_t)r * F4_PER_ROW + c4;
        float4 o;
        if (r < len) {
            const float4 v = x4[idx];
            o.x = fmaf((v.x - mean) * inv, w4.x, a4.x);
            o.y = fmaf((v.y - mean) * inv, w4.y, a4.y);
            o.z = fmaf((v.z - mean) * inv, w4.z, a4.z);
            o.w = fmaf((v.w - mean) * inv, w4.w, a4.w);
        } else {
            o.x = 0.0f; o.y = 0.0f; o.z = 0.0f; o.w = 0.0f;
        }
        o4[idx] = o;                               // invalid rows must be 0
    }
}

// ---------------------------------------------------------------- launch
extern "C" void kernel_launch(void* const* d_in, const int* in_sizes, int n_in,
                              void* d_out, int out_size, void* d_ws, size_t ws_size,
                              hipStream_t stream) {
    const float* x       = (const float*)d_in[0];
    const int*   lengths = (const int*)  d_in[1];
    const float* weights = (const float*)d_in[2];
    const float* biases  = (const float*)d_in[3];
    float*       out     = (float*)d_out;
    float*       ws      = (float*)d_ws;       // [0,64): sum  [64,128): sumsq

    setnorm_zero_ws<<<1, 2 * BATCH, 0, stream>>>(ws);
    dim3 grid(CHUNKS, BATCH);
    setnorm_reduce<<<grid, THREADS, 0, stream>>>(x, lengths, ws, ws + BATCH);
    setnorm_apply<<<grid, THREADS, 0, stream>>>(x, lengths, weights, biases,
                                                ws, ws + BATCH, out);
}